// Model_20624432955446
// MI455X (gfx1250) — compile-verified
//
#include <hip/hip_runtime.h>
#include <hip/hip_bf16.h>

// Cross product along dim=1 of (16, 3, 1024, 1024) float32 tensors.
// Pure streaming op: optimal MI455X strategy is b128 non-temporal
// loads/stores saturating the 23.3 TB/s HBM path. No matrix math.

typedef __attribute__((ext_vector_type(4))) float v4f;

// One component plane = 1024*1024 floats = 2^18 float4 groups.
#define PLANE_V4_LOG2 18
#define PLANE_V4      (1u << PLANE_V4_LOG2)

__global__ __launch_bounds__(256)
void cross_dim1_kernel(const float* __restrict__ A,
                       const float* __restrict__ B,
                       float* __restrict__ O,
                       unsigned int nGroups) {
    unsigned int g = blockIdx.x * blockDim.x + threadIdx.x;
    if (g >= nGroups) return;

    unsigned int batch = g >> PLANE_V4_LOG2;          // 0..15
    unsigned int pix   = g & (PLANE_V4 - 1u);         // float4 index in plane

    // Base offset (in float4 units) of the x-component plane for this batch.
    size_t base = (size_t)batch * (3u * PLANE_V4) + pix;

    const v4f* A4 = (const v4f*)A;
    const v4f* B4 = (const v4f*)B;
    v4f*       O4 = (v4f*)O;

    // Streamed once: non-temporal loads (TH=NT) keep the 576MB stream from
    // thrashing L2; four pixels per lane -> global_load_b128.
    v4f ax = __builtin_nontemporal_load(A4 + base);
    v4f ay = __builtin_nontemporal_load(A4 + base + PLANE_V4);
    v4f az = __builtin_nontemporal_load(A4 + base + 2u * PLANE_V4);
    v4f bx = __builtin_nontemporal_load(B4 + base);
    v4f by = __builtin_nontemporal_load(B4 + base + PLANE_V4);
    v4f bz = __builtin_nontemporal_load(B4 + base + 2u * PLANE_V4);

    // c = a x b (elementwise over the 4 packed pixels)
    v4f ox = ay * bz - az * by;
    v4f oy = az * bx - ax * bz;
    v4f oz = ax * by - ay * bx;

    __builtin_nontemporal_store(ox, O4 + base);
    __builtin_nontemporal_store(oy, O4 + base + PLANE_V4);
    __builtin_nontemporal_store(oz, O4 + base + 2u * PLANE_V4);
}

extern "C" void kernel_launch(void* const* d_in, const int* in_sizes, int n_in,
                              void* d_out, int out_size, void* d_ws, size_t ws_size,
                              hipStream_t stream) {
    const float* A = (const float*)d_in[0];
    const float* B = (const float*)d_in[1];
    float* O = (float*)d_out;

    // in_sizes[0] = 16*3*1024*1024 = 50331648 floats.
    // Each thread handles 4 pixels x 3 components = 12 elements.
    unsigned int nGroups = (unsigned int)(in_sizes[0] / 12);  // 4,194,304

    const int block = 256;
    const int grid  = (int)((nGroups + block - 1) / block);   // 16384 blocks
    cross_dim1_kernel<<<grid, block, 0, stream>>>(A, B, O, nGroups);
}